// NCCLoss_12541304504657
// MI455X (gfx1250) — compile-verified
//
#include <hip/hip_runtime.h>

typedef __attribute__((ext_vector_type(2))) float v2f;
typedef __attribute__((ext_vector_type(8))) float v8f;

#define IMGS   16
#define H      512
#define W      512
#define PAD    4        // win//2 for 9x9
#define TDIM   16       // output tile is 16x16 (one WMMA D tile)
#define WAVES  4        // waves (tiles) per workgroup
#define TSTR   26       // padded LDS row stride (floats): 26n mod 64 distinct for n=0..15
#define TILES_PER_IMG ((H / TDIM) * (W / TDIM))   // 1024
#define NTILES (IMGS * TILES_PER_IMG)             // 16384
#define NWG    (NTILES / WAVES)                   // 4096

// band(k, m) = 1 if output index m receives halo index k (9-wide window)
__device__ __forceinline__ float bandf(int k, int m) {
    return (k >= m && k <= m + 8) ? 1.0f : 0.0f;
}

__global__ __launch_bounds__(WAVES * 32)
void ncc_main(const float* __restrict__ I, const float* __restrict__ J,
              float* __restrict__ partial) {
    // Wave-private staging of horizontal box sums, stored TRANSPOSED:
    // tmp[wave][q][col][haloRow], haloRow in [0,24)
    __shared__ float tmp[WAVES][5][TDIM][TSTR];
    __shared__ float wgred[WAVES];

    const int lane = threadIdx.x & 31;
    const int wave = threadIdx.x >> 5;
    const int tile = blockIdx.x * WAVES + wave;

    const int img = tile / TILES_PER_IMG;
    const int t2  = tile % TILES_PER_IMG;
    const int ty  = t2 / (W / TDIM);
    const int tx  = t2 % (W / TDIM);
    const int y0  = ty * TDIM;
    const int x0  = tx * TDIM;

    const float* Ib = I + (size_t)img * H * W;
    const float* Jb = J + (size_t)img * H * W;

    const int  n    = lane & 15;          // M for A-operands, N for B/D operands
    const int  koff = (lane >= 16) ? 2 : 0;   // K pair handled by this half-wave

    // ---------------- Stage 1: horizontal box sums via WMMA ----------------
    // Two 16-row tiles of the 24-row halo: t=0 -> halo rows 0..15, t=1 -> 8..23
    #pragma unroll
    for (int t = 0; t < 2; ++t) {
        const int hrow = n + 8 * t;           // A-matrix row M = n
        const int y    = y0 - PAD + hrow;
        const bool yok = (y >= 0) && (y < H);

        v2f pi[6], pj[6];
        #pragma unroll
        for (int j = 0; j < 6; ++j) {
            const int x = x0 - PAD + 4 * j + koff;
            float i0 = 0.f, i1 = 0.f, ja = 0.f, jb = 0.f;
            if (yok) {
                const float* ri = Ib + (size_t)y * W;
                const float* rj = Jb + (size_t)y * W;
                if (x     >= 0 && x     < W) { i0 = ri[x];     ja = rj[x];     }
                if (x + 1 >= 0 && x + 1 < W) { i1 = ri[x + 1]; jb = rj[x + 1]; }
            }
            pi[j].x = i0; pi[j].y = i1;
            pj[j].x = ja; pj[j].y = jb;
        }

        #pragma unroll
        for (int q = 0; q < 5; ++q) {
            v8f acc = {};
            #pragma unroll
            for (int j = 0; j < 6; ++j) {
                v2f a;
                switch (q) {
                    case 0:  a = pi[j];         break;   // I
                    case 1:  a = pj[j];         break;   // J
                    case 2:  a = pi[j] * pi[j]; break;   // I*I
                    case 3:  a = pj[j] * pj[j]; break;   // J*J
                    default: a = pi[j] * pj[j]; break;   // I*J
                }
                const int kb = 4 * j + koff;
                v2f b; b.x = bandf(kb, n); b.y = bandf(kb + 1, n);
                acc = __builtin_amdgcn_wmma_f32_16x16x4_f32(
                          false, a, false, b, (short)0, acc, false, false);
            }
            // D layout: vgpr r -> row (r + 8*hi), col = n. Store transposed:
            const int rowbase = 8 * t + ((lane >= 16) ? 8 : 0);
            float* dst = &tmp[wave][q][n][rowbase];
            #pragma unroll
            for (int r = 0; r < 8; ++r) dst[r] = acc[r];
        }
    }
    // tmp was written and is read by the same wave; DS counter ordering suffices.

    // ---------------- Stage 2: vertical box sums via WMMA ----------------
    v8f s[5];
    #pragma unroll
    for (int q = 0; q < 5; ++q) {
        v8f acc = {};
        #pragma unroll
        for (int j = 0; j < 6; ++j) {
            const int kb = 4 * j + koff;
            v2f a; a.x = bandf(kb, n); a.y = bandf(kb + 1, n);  // BandV[m=n][k]
            v2f b;                                              // tmp[haloRow=k][col=n]
            b.x = tmp[wave][q][n][kb];
            b.y = tmp[wave][q][n][kb + 1];
            acc = __builtin_amdgcn_wmma_f32_16x16x4_f32(
                      false, a, false, b, (short)0, acc, false, false);
        }
        s[q] = acc;
    }

    // ---------------- Elementwise cc + reduction ----------------
    const float wsz = 81.0f;
    const float inv = 1.0f / 81.0f;
    float lacc = 0.f;
    #pragma unroll
    for (int r = 0; r < 8; ++r) {
        const float Is  = s[0][r], Js  = s[1][r];
        const float I2s = s[2][r], J2s = s[3][r], IJs = s[4][r];
        const float uI = Is * inv, uJ = Js * inv;
        const float cross = IJs - uJ * Is - uI * Js + uI * uJ * wsz;
        const float Ivar  = I2s - 2.f * uI * Is + uI * uI * wsz;
        const float Jvar  = J2s - 2.f * uJ * Js + uJ * uJ * wsz;
        lacc += cross * cross / (Ivar * Jvar + 1e-5f);
    }

    // wave32 reduce
    #pragma unroll
    for (int off = 16; off > 0; off >>= 1)
        lacc += __shfl_xor(lacc, off, 32);
    if (lane == 0) wgred[wave] = lacc;
    __syncthreads();
    if (threadIdx.x == 0) {
        float t = 0.f;
        #pragma unroll
        for (int wv = 0; wv < WAVES; ++wv) t += wgred[wv];
        partial[blockIdx.x] = t;
    }
}

__global__ __launch_bounds__(256)
void ncc_finish(const float* __restrict__ partial, float* __restrict__ out) {
    __shared__ float red[256];
    float s = 0.f;
    for (int i = threadIdx.x; i < NWG; i += 256) s += partial[i];
    red[threadIdx.x] = s;
    __syncthreads();
    for (int stride = 128; stride > 0; stride >>= 1) {
        if (threadIdx.x < stride) red[threadIdx.x] += red[threadIdx.x + stride];
        __syncthreads();
    }
    if (threadIdx.x == 0)
        out[0] = 1.0f - red[0] / ((float)IMGS * (float)H * (float)W);
}

extern "C" void kernel_launch(void* const* d_in, const int* in_sizes, int n_in,
                              void* d_out, int out_size, void* d_ws, size_t ws_size,
                              hipStream_t stream) {
    const float* I = (const float*)d_in[0];
    const float* J = (const float*)d_in[1];
    float* out     = (float*)d_out;
    float* partial = (float*)d_ws;   // NWG floats = 16 KB of scratch

    ncc_main<<<NWG, WAVES * 32, 0, stream>>>(I, J, partial);
    ncc_finish<<<1, 256, 0, stream>>>(partial, out);
}